// SpatioTemporalLayer1_15350213116360
// MI455X (gfx1250) — compile-verified
//
#include <hip/hip_runtime.h>
#include <math.h>

// ---------------------------------------------------------------------------
// SpatioTemporalLayer for MI455X (gfx1250, wave32, WMMA, async-to-LDS).
// All dense contractions run on V_WMMA_F32_16X16X4_F32 (fp32 matrix pipe).
// Shared operand panels are staged in LDS with GLOBAL_LOAD_ASYNC_TO_LDS_B128
// (ASYNCcnt-tracked CDNA5 async DMA path), exploiting the 320KB/WGP LDS.
// ---------------------------------------------------------------------------

typedef __attribute__((ext_vector_type(2))) float v2f;
typedef __attribute__((ext_vector_type(8))) float v8f;

#define LDIM 512
#define NDIM 64
#define DDIM 256
#define HEADS 8
#define HDIM 32
#define MTOT (LDIM * NDIM)          // 32768 tokens
#define ROWSTRIDE (NDIM * DDIM)     // 16384 floats between consecutive l (fixed n)

#define APITCH 260                  // LDS pitch for 256-wide panels (4 mod 64 banks)
#define SPITCH 516                  // LDS pitch for 64x512 score panel
#define QPITCH 36                   // LDS pitch for 32-wide head panels

__device__ __forceinline__ v8f wmma_f32(v2f a, v2f b, v8f c) {
  // 8-arg form: (neg_a, A, neg_b, B, c_mod, C, reuse_a, reuse_b)
  return __builtin_amdgcn_wmma_f32_16x16x4_f32(false, a, false, b, (short)0, c,
                                               false, false);
}

// LDS byte offset of a generic pointer into shared memory: the flat LDS
// aperture maps addr[31:0] directly to the LDS offset.
__device__ __forceinline__ unsigned lds_offset(const void* p) {
  return (unsigned)(unsigned long long)p;
}

// GLOBAL_LOAD_ASYNC_TO_LDS_B128: 16B memory -> 16B LDS per lane, ASYNCcnt.
__device__ __forceinline__ void async_load_b128(unsigned lds_off,
                                                const float* gptr) {
  asm volatile("global_load_async_to_lds_b128 %0, %1, off"
               :
               : "v"(lds_off), "v"((unsigned long long)gptr)
               : "memory");
}

__device__ __forceinline__ void wait_async() {
  asm volatile("s_wait_asynccnt 0x0" ::: "memory");
}

// ---------------------------------------------------------------------------
// Y[m, o] = sum_k X[m, k] * W[o, k] + bias[o]      (M x 256 x 256)
// grid.x = M/32 (32 rows per block), 256 threads = 8 waves.
// The 32x256 A panel is async-DMA'd to LDS once and shared by all 8 waves;
// each wave computes a 32x32 output block (2x2 grid of 16x16 WMMA tiles):
// 4 operand loads -> 4 WMMAs per k-step.
// ---------------------------------------------------------------------------
__global__ __launch_bounds__(256) void gemm256_bias(
    const float* __restrict__ X, const float* __restrict__ W,
    const float* __restrict__ bias, float* __restrict__ Y) {
  __shared__ float As[32 * APITCH];           // 33,280 B

  const int tid  = threadIdx.x;
  const int lane = tid & 31;
  const int wv   = tid >> 5;                  // 0..7
  const int m0   = blockIdx.x * 32;

  // --- async stage of A panel: 32 rows x 256 floats ---
#pragma unroll
  for (int i = 0; i < 8; ++i) {
    const int flat = i * 256 + tid;           // 0..2047 float4 slots
    const int r = flat >> 6, c4 = flat & 63;
    async_load_b128(lds_offset(As + r * APITCH + c4 * 4),
                    X + (size_t)(m0 + r) * DDIM + c4 * 4);
  }
  wait_async();
  __syncthreads();

  const int kq   = (lane >> 4) * 2;           // half-wave K offset
  const int col0 = wv * 32 + (lane & 15);     // B/C: lane&15 = N
  const int col1 = col0 + 16;
  const float* A0 = As + (lane & 15) * APITCH;          // rows m0..m0+15
  const float* A1 = As + (16 + (lane & 15)) * APITCH;   // rows m0+16..m0+31
  const float* B0 = W + (size_t)col0 * DDIM;  // row o of W == column o of W^T
  const float* B1 = W + (size_t)col1 * DDIM;

  v8f c00 = {}, c01 = {}, c10 = {}, c11 = {};
  for (int kk = 0; kk < DDIM; kk += 4) {
    __builtin_prefetch(B0 + kk + 64, 0, 1);   // global_prefetch_b8
    v2f a0 = *(const v2f*)(A0 + kk + kq);     // ds_load_b64
    v2f a1 = *(const v2f*)(A1 + kk + kq);
    v2f b0 = *(const v2f*)(B0 + kk + kq);     // global_load_b64 (L2-resident W)
    v2f b1 = *(const v2f*)(B1 + kk + kq);
    c00 = wmma_f32(a0, b0, c00);
    c01 = wmma_f32(a0, b1, c01);
    c10 = wmma_f32(a1, b0, c10);
    c11 = wmma_f32(a1, b1, c11);
  }
  const float bv0 = bias[col0];
  const float bv1 = bias[col1];
#pragma unroll
  for (int i = 0; i < 8; ++i) {               // C: VGPR i -> M = i + 8*(lane>=16)
    const int r0 = m0 + i + 8 * (lane >> 4);
    const int r1 = r0 + 16;
    Y[(size_t)r0 * DDIM + col0] = c00[i] + bv0;
    Y[(size_t)r0 * DDIM + col1] = c01[i] + bv1;
    Y[(size_t)r1 * DDIM + col0] = c10[i] + bv0;
    Y[(size_t)r1 * DDIM + col1] = c11[i] + bv1;
  }
}

// ---------------------------------------------------------------------------
// Per l: async-stage the 64x256 se/te slices in LDS, cosine-normalize rows
// in place, form both Grams ss = Us Us^T, ts = Ut Ut^T with WMMA (operands
// from LDS), and reduce w[l,n] = mean_j(ss[n,j] * ts[n,j]) via LDS atomics.
// grid = 512 (one l per block), 512 threads = 16 waves (one 16x16 tile each).
// dyn LDS = (2*64*260 + 64)*4 = 133,376 B.
// ---------------------------------------------------------------------------
__global__ __launch_bounds__(512) void gram_weight(
    const float* __restrict__ SE, const float* __restrict__ TE,
    float* __restrict__ Wout) {
  extern __shared__ float smem[];
  float* Us   = smem;                     // 64 x APITCH
  float* Ut   = smem + 64 * APITCH;       // 64 x APITCH
  float* wacc = smem + 2 * 64 * APITCH;   // 64

  const int tid = threadIdx.x;
  const int l   = blockIdx.x;
  if (tid < 64) wacc[tid] = 0.0f;

  // --- async stage: 64 rows x 256 floats, both panels ---
#pragma unroll
  for (int i = 0; i < 8; ++i) {
    const int flat = i * 512 + tid;       // 0..4095 float4 slots
    const int r = flat >> 6, c4 = flat & 63;
    const size_t g = ((size_t)l * NDIM + r) * DDIM + c4 * 4;
    async_load_b128(lds_offset(Us + r * APITCH + c4 * 4), SE + g);
    async_load_b128(lds_offset(Ut + r * APITCH + c4 * 4), TE + g);
  }
  wait_async();
  __syncthreads();

  // --- normalize rows in LDS: 8 threads per row, 32 floats each ---
  const int row = tid >> 3;               // 0..63
  const int sub = tid & 7;
  float ss = 0.f, st = 0.f;
#pragma unroll
  for (int c = 0; c < 8; ++c) {
    const int d = (sub + c * 8) * 4;
    float4 vs = *(const float4*)(Us + row * APITCH + d);
    float4 vt = *(const float4*)(Ut + row * APITCH + d);
    ss += vs.x * vs.x + vs.y * vs.y + vs.z * vs.z + vs.w * vs.w;
    st += vt.x * vt.x + vt.y * vt.y + vt.z * vt.z + vt.w * vt.w;
  }
#pragma unroll
  for (int m = 4; m >= 1; m >>= 1) {      // reduce within the 8-lane row group
    ss += __shfl_xor(ss, m, 8);
    st += __shfl_xor(st, m, 8);
  }
  const float invS = 1.0f / fmaxf(sqrtf(ss), 1e-8f);
  const float invT = 1.0f / fmaxf(sqrtf(st), 1e-8f);
#pragma unroll
  for (int c = 0; c < 8; ++c) {
    const int d = (sub + c * 8) * 4;
    float4 vs = *(const float4*)(Us + row * APITCH + d);
    float4 vt = *(const float4*)(Ut + row * APITCH + d);
    vs.x *= invS; vs.y *= invS; vs.z *= invS; vs.w *= invS;
    vt.x *= invT; vt.y *= invT; vt.z *= invT; vt.w *= invT;
    *(float4*)(Us + row * APITCH + d) = vs;
    *(float4*)(Ut + row * APITCH + d) = vt;
  }
  __syncthreads();

  // --- dual Gram via WMMA from LDS ---
  const int lane = tid & 31;
  const int wv   = tid >> 5;              // 0..15
  const int mt   = wv >> 2, nt = wv & 3;  // 4x4 tiles of 16x16
  const int kq   = (lane >> 4) * 2;
  const float* Am = Us + (mt * 16 + (lane & 15)) * APITCH;
  const float* Bm = Us + (nt * 16 + (lane & 15)) * APITCH;
  const float* At = Ut + (mt * 16 + (lane & 15)) * APITCH;
  const float* Bt = Ut + (nt * 16 + (lane & 15)) * APITCH;
  v8f cs = {};
  v8f ct = {};
  for (int kk = 0; kk < DDIM; kk += 4) {
    v2f a_s = *(const v2f*)(Am + kk + kq);
    v2f b_s = *(const v2f*)(Bm + kk + kq);
    cs = wmma_f32(a_s, b_s, cs);
    v2f a_t = *(const v2f*)(At + kk + kq);
    v2f b_t = *(const v2f*)(Bt + kk + kq);
    ct = wmma_f32(a_t, b_t, ct);
  }
#pragma unroll
  for (int i = 0; i < 8; ++i) {
    const int r = mt * 16 + i + 8 * (lane >> 4);
    atomicAdd(&wacc[r], cs[i] * ct[i]);   // ds_add_f32
  }
  __syncthreads();
  if (tid < 64) Wout[(size_t)l * NDIM + tid] = wacc[tid] * (1.0f / 64.0f);
}

// ---------------------------------------------------------------------------
// Attention for one (n, h, 64-query tile). Q (64x32), K (512x32), V (512x32)
// panels are async-DMA'd to LDS; the full 64x512 score panel lives in LDS.
//   Phase 1: S = scale * Q K^T        (WMMA, operands from LDS)
//   Phase 2: in-LDS row softmax       (v_exp_f32)
//   Phase 3: Ctx = P @ V              (WMMA, operands from LDS)
// grid = (8, H, N), 256 threads = 8 waves.
// dyn LDS = (64*516 + 64*36 + 2*512*36)*4 = 288,768 B (CDNA5: 320KB/WGP).
// ---------------------------------------------------------------------------
__global__ __launch_bounds__(256) void attention(
    const float* __restrict__ Q, const float* __restrict__ K,
    const float* __restrict__ V, float* __restrict__ Ctx) {
  extern __shared__ float smem[];
  float* sS = smem;                          // 64 x SPITCH
  float* Qs = smem + 64 * SPITCH;            // 64 x QPITCH
  float* Ks = Qs + 64 * QPITCH;              // 512 x QPITCH
  float* Vs = Ks + 512 * QPITCH;             // 512 x QPITCH

  const int tid   = threadIdx.x;
  const int lane  = tid & 31;
  const int wv    = tid >> 5;
  const int l0    = blockIdx.x * 64;
  const int h     = blockIdx.y;
  const int n     = blockIdx.z;
  const int qbase = n * DDIM + h * HDIM;     // (n,h) slice offset in a token row
  const int kq    = (lane >> 4) * 2;

  // --- async stage of Q / K / V head panels ---
#pragma unroll
  for (int i = 0; i < 2; ++i) {              // 64 rows x 8 float4
    const int flat = i * 256 + tid;
    const int r = flat >> 3, d4 = flat & 7;
    async_load_b128(lds_offset(Qs + r * QPITCH + d4 * 4),
                    Q + (size_t)(l0 + r) * ROWSTRIDE + qbase + d4 * 4);
  }
#pragma unroll
  for (int i = 0; i < 16; ++i) {             // 512 rows x 8 float4, K and V
    const int flat = i * 256 + tid;
    const int r = flat >> 3, d4 = flat & 7;
    const size_t g = (size_t)r * ROWSTRIDE + qbase + d4 * 4;
    async_load_b128(lds_offset(Ks + r * QPITCH + d4 * 4), K + g);
    async_load_b128(lds_offset(Vs + r * QPITCH + d4 * 4), V + g);
  }
  wait_async();
  __syncthreads();

  // --- Phase 1: scores (4 m-tiles x 32 key-tiles, LDS operands) ---
  for (int t = wv; t < 128; t += 8) {
    const int mt = t & 3;
    const int kt = t >> 2;
    const float* Aq = Qs + (mt * 16 + (lane & 15)) * QPITCH;
    const float* Bk = Ks + (kt * 16 + (lane & 15)) * QPITCH;
    v8f acc = {};
#pragma unroll
    for (int kk = 0; kk < HDIM; kk += 4) {
      v2f a = *(const v2f*)(Aq + kk + kq);
      v2f b = *(const v2f*)(Bk + kk + kq);
      acc = wmma_f32(a, b, acc);
    }
    const float scale = 0.17677669529663687f;  // 1/sqrt(32)
#pragma unroll
    for (int i = 0; i < 8; ++i)
      sS[(mt * 16 + i + 8 * (lane >> 4)) * SPITCH + kt * 16 + (lane & 15)] =
          acc[i] * scale;
  }
  __syncthreads();

  // --- Phase 2: softmax over 512 keys, one wave per row, 16 cols/lane ---
  for (int r = wv; r < 64; r += 8) {
    float* rowp = sS + r * SPITCH + lane * 16;
    float m = -3.4e38f;
#pragma unroll
    for (int j = 0; j < 16; ++j) m = fmaxf(m, rowp[j]);
    for (int msk = 16; msk >= 1; msk >>= 1) m = fmaxf(m, __shfl_xor(m, msk));
    float s = 0.f;
#pragma unroll
    for (int j = 0; j < 16; ++j) {
      const float e = __expf(rowp[j] - m);
      rowp[j] = e;
      s += e;
    }
    for (int msk = 16; msk >= 1; msk >>= 1) s += __shfl_xor(s, msk);
    const float inv = 1.0f / s;
#pragma unroll
    for (int j = 0; j < 16; ++j) rowp[j] *= inv;
  }
  __syncthreads();

  // --- Phase 3: Ctx = P @ V, 8 tiles (4 m x 2 d), one per wave ---
  {
    const int mt   = wv >> 1;
    const int nt   = wv & 1;
    const int dcol = nt * 16 + (lane & 15);
    const float* Ap = sS + (mt * 16 + (lane & 15)) * SPITCH;
    v8f acc = {};
    for (int kk = 0; kk < LDIM; kk += 4) {
      v2f a = *(const v2f*)(Ap + kk + kq);
      v2f b;                                  // B[k][d] = Vs[k][d]
      b.x = Vs[(kk + kq) * QPITCH + dcol];
      b.y = Vs[(kk + kq + 1) * QPITCH + dcol];
      acc = wmma_f32(a, b, acc);
    }
#pragma unroll
    for (int i = 0; i < 8; ++i)
      Ctx[(size_t)(l0 + mt * 16 + i + 8 * (lane >> 4)) * ROWSTRIDE + qbase +
          dcol] = acc[i];
  }
}

// ---------------------------------------------------------------------------
// out = x + w[token] * interaction
// ---------------------------------------------------------------------------
__global__ __launch_bounds__(256) void combine(
    const float* __restrict__ x, const float* __restrict__ wgt,
    const float* __restrict__ inter, float* __restrict__ out) {
  const size_t idx = ((size_t)blockIdx.x * 256 + threadIdx.x) * 4;
  const size_t t   = idx >> 8;            // /256 -> token index (l*N+n)
  const float wv   = wgt[t];
  float4 xv = *(const float4*)(x + idx);
  float4 iv = *(const float4*)(inter + idx);
  float4 o;
  o.x = xv.x + wv * iv.x;
  o.y = xv.y + wv * iv.y;
  o.z = xv.z + wv * iv.z;
  o.w = xv.w + wv * iv.w;
  *(float4*)(out + idx) = o;
}

// ---------------------------------------------------------------------------
extern "C" void kernel_launch(void* const* d_in, const int* in_sizes, int n_in,
                              void* d_out, int out_size, void* d_ws,
                              size_t ws_size, hipStream_t stream) {
  const float* x    = (const float*)d_in[0];
  const float* sp   = (const float*)d_in[1];
  const float* tp   = (const float*)d_in[2];
  const float* Ws   = (const float*)d_in[3];
  const float* bs   = (const float*)d_in[4];
  const float* Wt   = (const float*)d_in[5];
  const float* bt   = (const float*)d_in[6];
  const float* Wqkv = (const float*)d_in[7];
  const float* bqkv = (const float*)d_in[8];
  const float* Wo   = (const float*)d_in[9];
  const float* bo   = (const float*)d_in[10];
  float* out = (float*)d_out;

  // Scratch: 5 token-major [32768,256] fp32 buffers + w (32768) = ~168 MB.
  float* ws = (float*)d_ws;
  const size_t MD = (size_t)MTOT * DDIM;
  float* se  = ws;
  float* te  = ws + MD;
  float* q   = ws + 2 * MD;
  float* k   = ws + 3 * MD;
  float* v   = ws + 4 * MD;
  float* wgt = ws + 5 * MD;
  float* ctx   = se;  // se dead after gram_weight + q projection
  float* inter = te;  // te dead after gram_weight + k/v projections

  const dim3 blk(256);
  const int gemm_grid = MTOT / 32;  // 1024

  gemm256_bias<<<gemm_grid, blk, 0, stream>>>(sp, Ws, bs, se);
  gemm256_bias<<<gemm_grid, blk, 0, stream>>>(tp, Wt, bt, te);
  gemm256_bias<<<gemm_grid, blk, 0, stream>>>(se, Wqkv, bqkv, q);
  gemm256_bias<<<gemm_grid, blk, 0, stream>>>(te, Wqkv + DDIM * DDIM, bqkv + DDIM, k);
  gemm256_bias<<<gemm_grid, blk, 0, stream>>>(te, Wqkv + 2 * DDIM * DDIM,
                                              bqkv + 2 * DDIM, v);

  gram_weight<<<LDIM, 512, (2 * 64 * APITCH + 64) * sizeof(float), stream>>>(
      se, te, wgt);

  attention<<<dim3(LDIM / 64, HEADS, NDIM), blk,
              (64 * SPITCH + 64 * QPITCH + 2 * 512 * QPITCH) * sizeof(float),
              stream>>>(q, k, v, ctx);

  gemm256_bias<<<gemm_grid, blk, 0, stream>>>(ctx, Wo, bo, inter);

  combine<<<(MTOT * DDIM / 4) / 256, blk, 0, stream>>>(x, wgt, inter, out);
}